// QuantizedOutliers_47639777247712
// MI455X (gfx1250) — compile-verified
//
#include <hip/hip_runtime.h>
#include <stdint.h>

// ---------------------------------------------------------------------------
// QuantizedOutliers: escape-coded delta decode + int4 block dequant + scatter
// into zeroed 8192x8192 f32 dense output.
//
//   K0 zero_out : vectorized zero-fill of 256MB output  (bandwidth bound)
//   K1 pass1    : chunk records (3 entry states) + group record fold (LDS)
//   K2 pass2    : serial combine over ~43 GROUP records, all staged in LDS
//   K3 pass3    : group entry -> per-chunk entries (LDS) -> decode + scatter
//
// CDNA5 path: byte stream staged global->LDS with async B128 copies
// (ASYNCcnt / s_wait_asynccnt), decoded byte-wise out of LDS.
// ---------------------------------------------------------------------------

constexpr int CHUNK  = 256;            // bytes per decode chunk / thread
constexpr int TPB    = 64;             // threads per decode block (2 waves)
constexpr int GROUP  = TPB;            // chunks per group (== one block)
constexpr int GBYTES = CHUNK * GROUP;  // 16 KB byte window per group
constexpr int NTILE  = 1024;           // group records staged per K2 tile

// Vector type matching the async-LDS builtin's expected pointee (v4i32).
typedef int v4i __attribute__((vector_size(16)));
typedef __attribute__((address_space(1))) v4i glb_v4i;
typedef __attribute__((address_space(3))) v4i lds_v4i;

// Stage [blockStart, blockStart+nbytes) of the byte stream into LDS.
// All TPB threads must call this (contains barrier).
__device__ __forceinline__ void stage_to_lds(uint8_t* lds, const int8_t* bytes,
                                             long long blockStart, int nbytes,
                                             int tid) {
    int full16 = nbytes >> 4;  // whole 16B granules
#if __has_builtin(__builtin_amdgcn_global_load_async_to_lds_b128)
    for (int i = tid; i < full16; i += TPB) {
        __builtin_amdgcn_global_load_async_to_lds_b128(
            (glb_v4i*)(bytes + blockStart + (long long)i * 16),
            (lds_v4i*)(lds + i * 16),
            /*offset=*/0, /*cpol=*/0);
    }
    for (int i = (full16 << 4) + tid; i < nbytes; i += TPB)   // tail (<16B)
        lds[i] = (uint8_t)bytes[blockStart + i];
  #if __has_builtin(__builtin_amdgcn_s_wait_asynccnt)
    __builtin_amdgcn_s_wait_asynccnt(0);
  #else
    asm volatile("s_wait_asynccnt 0" ::: "memory");
  #endif
#else
    for (int i = tid; i < full16; i += TPB)
        ((int4*)lds)[i] = ((const int4*)(bytes + blockStart))[i];
    for (int i = (full16 << 4) + tid; i < nbytes; i += TPB)
        lds[i] = (uint8_t)bytes[blockStart + i];
#endif
    __syncthreads();
}

// int4 block dequant: vals[r] = mn[r/64] + (mx-mn)[r/64] * q[r]/15
__device__ __forceinline__ float dequant(uint32_t r, const int8_t* packed,
                                         const float* mn, const float* mx) {
    int up = ((int)(uint8_t)packed[r >> 1]) ^ 0x80;   // packed+128 in [0,255]
    int q  = (r & 1u) ? (up & 15) : (up >> 4);
    uint32_t b = r >> 6;
    float lo = mn[b], hi = mx[b];
    return lo + (hi - lo) * ((float)q / 15.0f);
}

// ---------------------------------------------------------------------------
// K0: zero the dense output (float4 grid-stride, scalar tail).
// ---------------------------------------------------------------------------
__global__ void zero_out_kernel(float* __restrict__ out, long long n) {
    long long n4     = n >> 2;
    long long i      = (long long)blockIdx.x * blockDim.x + threadIdx.x;
    long long stride = (long long)gridDim.x * blockDim.x;
    float4 z = make_float4(0.f, 0.f, 0.f, 0.f);
    float4* o4 = (float4*)out;
    for (long long k = i; k < n4; k += stride) o4[k] = z;
    if (i == 0)
        for (long long k = n4 << 2; k < n; ++k) out[k] = 0.f;
}

// ---------------------------------------------------------------------------
// Automaton: state 0 = free, 2 = expect hi byte, 1 = expect lo byte.
// A chunk record maps entry state -> (exit state, delta-sum, valid-count);
// sums for entry state 1 are computed with hi=0, the +hi*256 correction is
// applied by whoever knows the previous byte (associative composition).
// ---------------------------------------------------------------------------

// K1: per-chunk records + group-record fold.
// rec layout  per chunk : [map0..2, sum0..2, cnt0..2]
// grec layout per group : same, composed over the group's chunks.
__global__ void pass1_kernel(const int8_t* __restrict__ bytes, long long B,
                             int nchunk, uint32_t* __restrict__ rec,
                             uint32_t* __restrict__ grec) {
    __shared__ __align__(16) uint8_t sbytes[GBYTES];
    __shared__ uint32_t lmap[GROUP][3], lsum[GROUP][3], lcnt[GROUP][3];

    int g = blockIdx.x;
    long long blockStart = (long long)g * GBYTES;
    long long rem = B - blockStart;
    int navail = (rem <= 0) ? 0 : (rem < (long long)GBYTES ? (int)rem : GBYTES);
    stage_to_lds(sbytes, bytes, blockStart, navail, threadIdx.x);  // barrier

    int lc = threadIdx.x;
    int c  = g * GROUP + lc;
    if (c < nchunk) {
        int off = lc * CHUNK;
        long long cstart = (long long)c * CHUNK;
        int len = (B - cstart < (long long)CHUNK) ? (int)(B - cstart) : CHUNK;
        for (int s0 = 0; s0 < 3; ++s0) {
            int state = s0, hi = 0;
            uint32_t sum = 0, cnt = 0;
            for (int i = 0; i < len; ++i) {
                int u = ((int)sbytes[off + i]) ^ 0x80;   // int8 + 128
                if (state == 0) {
                    if (u == 0) state = 2;               // escape
                    else { sum += (uint32_t)u; cnt++; }
                } else if (state == 2) {
                    hi = u; state = 1;
                } else {                                  // state == 1
                    sum += (uint32_t)(hi * 256 + u); cnt++; state = 0;
                }
            }
            lmap[lc][s0] = (uint32_t)state;
            lsum[lc][s0] = sum;
            lcnt[lc][s0] = cnt;
            size_t base = (size_t)c * 9;
            rec[base + 0 + s0] = (uint32_t)state;
            rec[base + 3 + s0] = sum;
            rec[base + 6 + s0] = cnt;
        }
    }
    __syncthreads();

    // Fold the group's chunk records; one thread per entry state.
    int gcount = nchunk - g * GROUP;
    if (gcount > GROUP) gcount = GROUP;
    if (threadIdx.x < 3 && gcount > 0) {
        uint32_t state = threadIdx.x, sum = 0, cnt = 0;
        for (int j = 0; j < gcount; ++j) {
            uint32_t cs = lsum[j][state];
            if (state == 1 && j > 0)  // hi byte = last byte of previous chunk
                cs += (uint32_t)(((int)sbytes[j * CHUNK - 1]) ^ 0x80) * 256u;
            cnt += lcnt[j][state];
            uint32_t ns = lmap[j][state];
            sum += cs; state = ns;
        }
        size_t base = (size_t)g * 9;
        grec[base + 0 + threadIdx.x] = state;
        grec[base + 3 + threadIdx.x] = sum;
        grec[base + 6 + threadIdx.x] = cnt;
    }
}

// ---------------------------------------------------------------------------
// K2: the only serial step — combine group records. Records and group
// boundary bytes are staged cooperatively into LDS so thread 0's dependent
// chain is ~ngroup LDS reads, not global ones.
// gentry per group: [state, run_prefix, rank_prefix]
// ---------------------------------------------------------------------------
__global__ void pass2_kernel(const int8_t* __restrict__ bytes, int ngroup,
                             const uint32_t* __restrict__ grec,
                             uint32_t* __restrict__ gentry) {
    __shared__ uint32_t tmap[NTILE][3], tsum[NTILE][3], tcnt[NTILE][3];
    __shared__ uint8_t  tby[NTILE];

    uint32_t state = 0, run = 0, rank = 0;   // meaningful in thread 0 only
    for (int t0 = 0; t0 < ngroup; t0 += NTILE) {
        int tc = ngroup - t0;
        if (tc > NTILE) tc = NTILE;
        for (int j = threadIdx.x; j < tc; j += blockDim.x) {
            size_t base = (size_t)(t0 + j) * 9;
            for (int s = 0; s < 3; ++s) {
                tmap[j][s] = grec[base + 0 + s];
                tsum[j][s] = grec[base + 3 + s];
                tcnt[j][s] = grec[base + 6 + s];
            }
            long long bi = (long long)(t0 + j) * GBYTES - 1;
            tby[j] = (bi >= 0) ? (uint8_t)bytes[bi] : (uint8_t)0;
        }
        __syncthreads();
        if (threadIdx.x == 0) {
            for (int j = 0; j < tc; ++j) {
                int gg = t0 + j;
                gentry[(size_t)gg * 3 + 0] = state;
                gentry[(size_t)gg * 3 + 1] = run;
                gentry[(size_t)gg * 3 + 2] = rank;
                uint32_t sum = tsum[j][state];
                if (state == 1)  // hi byte = last byte before this group
                    sum += (uint32_t)(((int)tby[j]) ^ 0x80) * 256u;
                rank += tcnt[j][state];
                uint32_t ns = tmap[j][state];
                run += sum; state = ns;
            }
        }
        __syncthreads();
    }
}

// ---------------------------------------------------------------------------
// K3: expand group entry to per-chunk entries in LDS, then parallel decode
// + dequant + scatter.
// ---------------------------------------------------------------------------
__global__ void pass3_kernel(const int8_t* __restrict__ bytes, long long B,
                             int nchunk, const uint32_t* __restrict__ rec,
                             const uint32_t* __restrict__ gentry,
                             const long long* __restrict__ first_idx,
                             const int8_t* __restrict__ packed,
                             const float* __restrict__ mn,
                             const float* __restrict__ mx,
                             float* __restrict__ out) {
    __shared__ __align__(16) uint8_t sbytes[GBYTES];
    __shared__ uint32_t lmap[GROUP][3], lsum[GROUP][3], lcnt[GROUP][3];
    __shared__ uint32_t estate[GROUP], erun[GROUP], erank[GROUP];

    int g = blockIdx.x;
    long long blockStart = (long long)g * GBYTES;
    long long rem = B - blockStart;
    int navail = (rem <= 0) ? 0 : (rem < (long long)GBYTES ? (int)rem : GBYTES);
    stage_to_lds(sbytes, bytes, blockStart, navail, threadIdx.x);  // barrier

    int lc = threadIdx.x;
    int c  = g * GROUP + lc;
    if (c < nchunk) {
        size_t base = (size_t)c * 9;
        for (int s = 0; s < 3; ++s) {
            lmap[lc][s] = rec[base + 0 + s];
            lsum[lc][s] = rec[base + 3 + s];
            lcnt[lc][s] = rec[base + 6 + s];
        }
    }
    __syncthreads();

    int gcount = nchunk - g * GROUP;
    if (gcount > GROUP) gcount = GROUP;
    if (threadIdx.x == 0 && gcount > 0) {
        uint32_t state = gentry[(size_t)g * 3 + 0];
        uint32_t run   = gentry[(size_t)g * 3 + 1];
        uint32_t rank  = gentry[(size_t)g * 3 + 2];
        for (int j = 0; j < gcount; ++j) {
            estate[j] = state; erun[j] = run; erank[j] = rank;
            uint32_t sum = lsum[j][state];
            if (state == 1) {
                int hb = (j > 0)
                    ? (((int)sbytes[j * CHUNK - 1]) ^ 0x80)
                    : ((g > 0) ? (((int)(uint8_t)bytes[blockStart - 1]) ^ 0x80) : 0);
                sum += (uint32_t)hb * 256u;
            }
            rank += lcnt[j][state];
            uint32_t ns = lmap[j][state];
            run += sum; state = ns;
        }
    }
    __syncthreads();

    if (c >= nchunk) return;
    int idx0 = (int)first_idx[0];
    if (c == 0) out[idx0] = dequant(0, packed, mn, mx);   // vals[0] override

    int off = lc * CHUNK;
    long long cstart = (long long)c * CHUNK;
    int len = (B - cstart < (long long)CHUNK) ? (int)(B - cstart) : CHUNK;

    uint32_t state = estate[lc];
    uint32_t run   = erun[lc];
    uint32_t rank  = erank[lc];
    int hi = 0;
    if (state == 1)
        hi = (lc > 0) ? (((int)sbytes[off - 1]) ^ 0x80)
                      : (((int)(uint8_t)bytes[blockStart - 1]) ^ 0x80);

    for (int i = 0; i < len; ++i) {
        int u = ((int)sbytes[off + i]) ^ 0x80;
        if (state == 0) {
            if (u == 0) { state = 2; }
            else {
                run += (uint32_t)u; rank++;
                out[(uint32_t)idx0 + run] = dequant(rank, packed, mn, mx);
            }
        } else if (state == 2) {
            hi = u; state = 1;
        } else {  // state == 1
            run += (uint32_t)(hi * 256 + u); rank++; state = 0;
            out[(uint32_t)idx0 + run] = dequant(rank, packed, mn, mx);
        }
    }
}

// ---------------------------------------------------------------------------
extern "C" void kernel_launch(void* const* d_in, const int* in_sizes, int n_in,
                              void* d_out, int out_size, void* d_ws, size_t ws_size,
                              hipStream_t stream) {
    const long long* first_idx = (const long long*)d_in[0];
    const int8_t*    bytes     = (const int8_t*)d_in[1];
    const int8_t*    packed    = (const int8_t*)d_in[2];
    const float*     mn        = (const float*)d_in[3];
    const float*     mx        = (const float*)d_in[4];
    (void)n_in; (void)ws_size;

    long long B = (long long)in_sizes[1];
    int nchunk  = (int)((B + CHUNK - 1) / CHUNK);
    if (nchunk < 1) nchunk = 1;
    int ngroup  = (nchunk + GROUP - 1) / GROUP;

    uint32_t* rec    = (uint32_t*)d_ws;                    // nchunk * 9 u32
    uint32_t* grec   = rec  + (size_t)nchunk * 9;          // ngroup * 9 u32
    uint32_t* gentry = grec + (size_t)ngroup * 9;          // ngroup * 3 u32

    float* out = (float*)d_out;

    // K0: zero-fill 256MB output (bandwidth-dominant step)
    zero_out_kernel<<<dim3(4096), dim3(256), 0, stream>>>(out, (long long)out_size);

    pass1_kernel<<<dim3(ngroup), dim3(TPB), 0, stream>>>(bytes, B, nchunk, rec, grec);
    pass2_kernel<<<dim3(1), dim3(256), 0, stream>>>(bytes, ngroup, grec, gentry);
    pass3_kernel<<<dim3(ngroup), dim3(TPB), 0, stream>>>(
        bytes, B, nchunk, rec, gentry, first_idx, packed, mn, mx, out);
}